// ModuleCorrelation_5128190951773
// MI455X (gfx1250) — compile-verified
//
#include <hip/hip_runtime.h>

typedef __attribute__((ext_vector_type(2))) float v2f;
typedef __attribute__((ext_vector_type(8))) float v8f;

#define MAXD   3
#define NDISP  7            // 2*MAXD+1
#define NDISP2 49
#define C_CH   128
#define H_DIM  128
#define W_DIM  192
#define N_B    8
#define HW     (H_DIM * W_DIM)

// One block per (n, y): 12 waves, one 16-pixel x-tile per wave.
// Each wave accumulates ALL 7 dy at once: per K-step (4 channels) it loads
// one A fragment (tile pixels of `one`) and, per dy, two B fragments of `two`
// (columns x0-3..x0+12 and x0+5..x0+20), feeding 14 independent
// V_WMMA_F32_16X16X4_F32 chains:
//   D_left [i][j] = sum_c one[x0+i,c] * two[y+dy, x0-3+j, c]  -> i<8,  dx=j-3-i
//   D_right[i][j] = sum_c one[x0+i,c] * two[y+dy, x0+5+j, c]  -> i>=8, dx=j+5-i
// Out-of-range rows/columns are loaded clamped (memory safety) and the
// corresponding outputs overwritten with 0 at extraction (zero padding).
__global__ __launch_bounds__(384) void corr_wmma_kernel(
    const float* __restrict__ one,
    const float* __restrict__ two,
    float* __restrict__ out)
{
    __shared__ float lds[12][2][2][16][16];   // [wave][dy parity][L/R] = 48 KB

    const int y = blockIdx.x % H_DIM;
    const int n = blockIdx.x / H_DIM;

    const int wave    = threadIdx.x >> 5;
    const int lane    = threadIdx.x & 31;
    const int laneLow = lane & 15;
    const int laneHi  = lane >> 4;
    const int x0      = wave * 16;

    const float* oneRow = one + (size_t)n * C_CH * HW + (size_t)y * W_DIM;
    const float* twoPln = two + (size_t)n * C_CH * HW;

    // wave-uniform clamped row bases (SGPR-resident)
    const float* twoRowDy[NDISP];
    #pragma unroll
    for (int dyi = 0; dyi < NDISP; ++dyi) {
        int ty = y + dyi - MAXD;
        ty = ty < 0 ? 0 : (ty >= H_DIM ? H_DIM - 1 : ty);
        twoRowDy[dyi] = twoPln + (size_t)ty * W_DIM;
    }

    // lane-varying, loop-invariant element offsets
    // fragment mapping (wave32 fp32 16x4 A / 4x16 B):
    //   lane l -> M/N = l&15, channels k0 + 2*(l>>4) + {0,1}
    const int xa = x0 + laneLow;
    int xl = x0 - 3 + laneLow; xl = xl < 0 ? 0 : xl;                   // clamp
    int xr = x0 + 5 + laneLow; xr = xr >= W_DIM ? W_DIM - 1 : xr;      // clamp
    const int voffA = 2 * laneHi * HW + xa;
    const int voffL = 2 * laneHi * HW + xl;
    const int voffR = 2 * laneHi * HW + xr;

    const v8f vzero = {0.f,0.f,0.f,0.f,0.f,0.f,0.f,0.f};
    v8f accL[NDISP], accR[NDISP];
    #pragma unroll
    for (int d = 0; d < NDISP; ++d) { accL[d] = vzero; accR[d] = vzero; }

    for (int k0 = 0; k0 < C_CH; k0 += 4) {
        const size_t koff = (size_t)k0 * HW;

        v2f a;
        a.x = oneRow[koff + voffA];
        a.y = oneRow[koff + voffA + HW];

        #pragma unroll
        for (int dyi = 0; dyi < NDISP; ++dyi) {
            const float* pT = twoRowDy[dyi] + koff;
            v2f bl, br;
            bl.x = pT[voffL];
            bl.y = pT[voffL + HW];
            br.x = pT[voffR];
            br.y = pT[voffR + HW];
            accL[dyi] = __builtin_amdgcn_wmma_f32_16x16x4_f32(
                            false, a, false, bl, (short)0, accL[dyi], false, false);
            accR[dyi] = __builtin_amdgcn_wmma_f32_16x16x4_f32(
                            false, a, false, br, (short)0, accR[dyi], false, false);
        }
    }

    // Band-diagonal extraction through LDS (double-buffered on dy parity).
    // C/D layout: VGPR v, lanes 0-15 -> M=v, lanes 16-31 -> M=v+8; N = lane&15.
    const float invC = 1.0f / (float)C_CH;
    #pragma unroll
    for (int dyi = 0; dyi < NDISP; ++dyi) {
        const int p = dyi & 1;
        #pragma unroll
        for (int v = 0; v < 8; ++v) {
            lds[wave][p][0][v + 8 * laneHi][laneLow] = accL[dyi][v];
            lds[wave][p][1][v + 8 * laneHi][laneLow] = accR[dyi][v];
        }
        // same-wave cross-lane LDS RAW: drain DScnt before reading back
        asm volatile("s_wait_dscnt 0x0" ::: "memory");

        const int  dy    = dyi - MAXD;
        const bool tyOOB = (y + dy < 0) || (y + dy >= H_DIM);

        if (lane < 16) {
            const int i = lane;
            const int x = x0 + i;
            #pragma unroll
            for (int dxi = 0; dxi < NDISP; ++dxi) {
                const int dx = dxi - MAXD;
                const int xs = x + dx;
                float val;
                if (tyOOB || xs < 0 || xs >= W_DIM) {
                    val = 0.0f;                       // zero-pad region
                } else {
                    val = ((i < 8) ? lds[wave][p][0][i][i + dx + 3]
                                   : lds[wave][p][1][i][i + dx - 5]) * invC;
                }
                const int d = dyi * NDISP + dxi;
                out[(((size_t)n * NDISP2 + d) * H_DIM + y) * W_DIM + x] = val;
            }
        }
    }
}

extern "C" void kernel_launch(void* const* d_in, const int* in_sizes, int n_in,
                              void* d_out, int out_size, void* d_ws, size_t ws_size,
                              hipStream_t stream)
{
    const float* tenOne = (const float*)d_in[0];
    const float* tenTwo = (const float*)d_in[1];
    float* out = (float*)d_out;

    const int blocks = N_B * H_DIM;      // one block per (n, y) row = 1024
    corr_wmma_kernel<<<blocks, 384, 0, stream>>>(tenOne, tenTwo, out);
}